// ApplyAttentionPolicyMap_91104846283204
// MI455X (gfx1250) — compile-verified
//
#include <hip/hip_runtime.h>
#include <hip/hip_bf16.h>
#include <stdint.h>

// ---------------------------------------------------------------------------
// ApplyAttentionPolicyMap: out[B,1858] = concat(logits, pp_logits)[B,4288] @ fc1
// fc1 is a one-hot-per-column selection matrix -> the matmul is a gather:
//   out[b, j] = flat[b, sel[j]] * val[j]
// Memory-bound: ~233 MB of traffic -> ~10us at 23.3 TB/s, vs >100us for any
// WMMA dense-matmul formulation of the same 130.6 GFLOP. We therefore take the
// CDNA5 data-movement path: async global->LDS staging + LDS gather.
// ---------------------------------------------------------------------------

#define APM_IN    4288
#define APM_LOG   4096
#define APM_PP    192
#define APM_OUT   1858
#define APM_NROWS 2          // batch rows staged in LDS per block
#define APM_BLK   256        // 8 wave32 per block

#if defined(__AMDGCN__) && __has_builtin(__builtin_amdgcn_global_load_async_to_lds_b128)
#define APM_HAVE_ASYNC_LDS 1
typedef int apm_v4i __attribute__((vector_size(16)));
typedef __attribute__((address_space(1))) apm_v4i apm_glb_v4i;
typedef __attribute__((address_space(3))) apm_v4i apm_lds_v4i;
#else
#define APM_HAVE_ASYNC_LDS 0
#endif

__device__ __forceinline__ void apm_async_cp16(float* dst_lds, const float* src_glb) {
#if APM_HAVE_ASYNC_LDS
    __builtin_amdgcn_global_load_async_to_lds_b128(
        (apm_glb_v4i*)const_cast<float*>(src_glb),
        (apm_lds_v4i*)dst_lds,
        /*offset=*/0, /*cpol=*/0);
#else
    *(float4*)dst_lds = *(const float4*)src_glb;
#endif
}

// ---------------------------------------------------------------------------
// Stage 1a: init selection tables (d_ws is poisoned, make state deterministic)
// ---------------------------------------------------------------------------
__global__ void apm_sel_init_kernel(int* __restrict__ sel_idx,
                                    float* __restrict__ sel_val) {
    int j = blockIdx.x * blockDim.x + threadIdx.x;
    if (j < APM_OUT) {
        sel_idx[j] = 0;
        sel_val[j] = 0.0f;
    }
}

// ---------------------------------------------------------------------------
// Stage 1b: scan fc1 (row-major, coalesced across j) for the one nonzero per
// column. 2-D grid: x tiles columns, y tiles the 4288 rows in chunks of 134.
// Exactly one chunk finds the nonzero -> plain store, no race.
// ---------------------------------------------------------------------------
__global__ void apm_sel_scan_kernel(const float* __restrict__ fc1,
                                    int* __restrict__ sel_idx,
                                    float* __restrict__ sel_val) {
    int j = blockIdx.x * APM_BLK + threadIdx.x;
    if (j >= APM_OUT) return;
    const int i0 = blockIdx.y * 134;           // 134 * 32 == 4288 exactly
    int   found = -1;
    float fv    = 0.0f;
    const float* col = fc1 + (long)i0 * APM_OUT + j;
    for (int k = 0; k < 134; ++k) {
        float v = __builtin_nontemporal_load(col);
        col += APM_OUT;
        if (v != 0.0f) { found = i0 + k; fv = v; }
    }
    if (found >= 0) {
        sel_idx[j] = found;
        sel_val[j] = fv;
    }
}

// ---------------------------------------------------------------------------
// Stage 2: gather. Each block stages APM_NROWS concatenated rows into LDS via
// the CDNA5 async global->LDS copies (ASYNCcnt), then resolves 1858 outputs
// per row from LDS and streams them out with non-temporal stores.
// ---------------------------------------------------------------------------
__global__ __launch_bounds__(APM_BLK)
void apm_gather_kernel(const float* __restrict__ logits,
                       const float* __restrict__ pp,
                       const int* __restrict__ sel_idx,
                       const float* __restrict__ sel_val,
                       float* __restrict__ out) {
    __shared__ float rowbuf[APM_NROWS * APM_IN];   // 34304 B
    __shared__ int   s_idx[APM_OUT];               //  7432 B
    __shared__ float s_val[APM_OUT];               //  7432 B

    const int  tid = threadIdx.x;
    const long b0  = (long)blockIdx.x * APM_NROWS;

    const float* glog = logits + b0 * APM_LOG;
    const float* gpp  = pp     + b0 * APM_PP;

    // --- main logits payload: NROWS * 4096 floats = 2048 x 16B packets ---
    for (int v = tid; v < APM_NROWS * (APM_LOG / 4); v += APM_BLK) {
        const int r = v >> 10;               // /1024
        const int c = (v & 1023) << 2;       // float index, 16B aligned
        apm_async_cp16(&rowbuf[r * APM_IN + c], glog + (long)r * APM_LOG + c);
    }
    // --- pp_logits tail: NROWS * 192 floats = 96 x 16B packets ---
    for (int v = tid; v < APM_NROWS * (APM_PP / 4); v += APM_BLK) {
        const int r = v / (APM_PP / 4);
        const int c = (v % (APM_PP / 4)) << 2;
        apm_async_cp16(&rowbuf[r * APM_IN + APM_LOG + c], gpp + (long)r * APM_PP + c);
    }
    // --- selection tables (hot in L2, read once per block) ---
    for (int j = tid; j < APM_OUT; j += APM_BLK) {
        s_idx[j] = sel_idx[j];
        s_val[j] = sel_val[j];
    }

#if APM_HAVE_ASYNC_LDS
#if __has_builtin(__builtin_amdgcn_s_wait_asynccnt)
    __builtin_amdgcn_s_wait_asynccnt(0);
#else
    asm volatile("s_wait_asynccnt 0" ::: "memory");
#endif
#endif
    __syncthreads();

    // --- gather: random reads hit LDS banks, writes stream out NT ---
    float* orow = out + b0 * APM_OUT;
    for (int j = tid; j < APM_OUT; j += APM_BLK) {
        const int   idx = s_idx[j];
        const float vv  = s_val[j];
        float a0 = rowbuf[idx] * vv;
        float a1 = rowbuf[APM_IN + idx] * vv;
        __builtin_nontemporal_store(a0, orow + j);
        __builtin_nontemporal_store(a1, orow + APM_OUT + j);
    }
}

// ---------------------------------------------------------------------------
// Host launcher
// ---------------------------------------------------------------------------
extern "C" void kernel_launch(void* const* d_in, const int* in_sizes, int n_in,
                              void* d_out, int out_size, void* d_ws, size_t ws_size,
                              hipStream_t stream) {
    const float* logits = (const float*)d_in[0];   // [B, 64, 64] f32
    const float* pp     = (const float*)d_in[1];   // [B, 8, 24]  f32
    const float* fc1    = (const float*)d_in[2];   // [4288, 1858] f32
    float*       out    = (float*)d_out;           // [B, 1858]   f32

    const int B = in_sizes[0] / APM_LOG;           // 8192

    // scratch layout: sel_idx[1858] at 0, sel_val[1858] at +8192B
    int*   sel_idx = (int*)d_ws;
    float* sel_val = (float*)((char*)d_ws + 8192);

    apm_sel_init_kernel<<<(APM_OUT + APM_BLK - 1) / APM_BLK, APM_BLK, 0, stream>>>(
        sel_idx, sel_val);
    apm_sel_scan_kernel<<<dim3((APM_OUT + APM_BLK - 1) / APM_BLK, 32), APM_BLK, 0, stream>>>(
        fc1, sel_idx, sel_val);
    apm_gather_kernel<<<B / APM_NROWS, APM_BLK, 0, stream>>>(
        logits, pp, sel_idx, sel_val, out);
}